// GCN_block_67233418051652
// MI455X (gfx1250) — compile-verified
//
#include <hip/hip_runtime.h>

#define N_NODES 100000
#define N_EDGES 1280000
#define CH 64

typedef __attribute__((ext_vector_type(2))) float v2f;
typedef __attribute__((ext_vector_type(8))) float v8f;

// ---------------------------------------------------------------------------
// 1) deg[i] = 1.0 (self-loop)
__global__ __launch_bounds__(256) void k_init_deg(float* __restrict__ deg) {
  int i = blockIdx.x * 256 + threadIdx.x;
  if (i < N_NODES) deg[i] = 1.0f;
}

// 2) deg[dst[e]] += 1
__global__ __launch_bounds__(256) void k_count_deg(const int* __restrict__ ei,
                                                   float* __restrict__ deg) {
  int e = blockIdx.x * 256 + threadIdx.x;
  if (e < N_EDGES) atomicAdd(&deg[ei[N_EDGES + e]], 1.0f);
}

// 3) dinv[i] = rsqrt(deg[i])   (deg >= 1 always, self-loops guarantee it)
__global__ __launch_bounds__(256) void k_rsqrt(float* __restrict__ deg) {
  int i = blockIdx.x * 256 + threadIdx.x;
  if (i < N_NODES) deg[i] = rsqrtf(deg[i]);
}

// 4) hs = (x @ W) * dinv[row], acc(d_out) initialized to hs (self-loop term).
//    One wave computes one 16x16 f32 tile via V_WMMA_F32_16X16X4_F32, K=64.
__global__ __launch_bounds__(256) void k_gemm(const float* __restrict__ x,
                                              const float* __restrict__ W,
                                              const float* __restrict__ dinv,
                                              float* __restrict__ hs,
                                              float* __restrict__ acc) {
  const int lane = threadIdx.x & 31;
  const int wave = (blockIdx.x * 256 + threadIdx.x) >> 5;   // global wave id
  const int tile_m = wave >> 2;          // 6250 row tiles (100000/16, exact)
  const int n0     = (wave & 3) << 4;    // 4 col tiles of 16 (CH=64)

  const int mrow = tile_m * 16 + (lane & 15);
  const int koff = (lane >> 4) << 1;     // lanes 0-15: K={0,1}; 16-31: K={2,3}
  const int col  = n0 + (lane & 15);

  v8f c = {};
#pragma unroll
  for (int k = 0; k < CH; k += 4) {
    // A 16x4: float2 per lane, 8B-aligned contiguous load
    v2f a = *(const v2f*)(x + mrow * CH + k + koff);
    // B 4x16: row K striped across lanes; two strided scalar loads
    v2f b;
    b.x = W[(k + koff) * CH + col];
    b.y = W[(k + koff + 1) * CH + col];
    // 8 args: (neg_a, A, neg_b, B, c_mod, C, reuse_a, reuse_b)
    c = __builtin_amdgcn_wmma_f32_16x16x4_f32(false, a, false, b,
                                              (short)0, c, false, false);
  }

  // C/D layout: VGPR r holds M=r (lanes 0-15) and M=r+8 (lanes 16-31), N=lane&15
  const int rbase = (lane >> 4) << 3;
#pragma unroll
  for (int r = 0; r < 8; ++r) {
    int row = tile_m * 16 + rbase + r;
    float v = c[r] * dinv[row];
    hs[row * CH + col]  = v;   // read-only source for scatter
    acc[row * CH + col] = v;   // accumulator init = self-loop contribution
  }
}

// 5) acc[dst] += hs[src]; one wave per edge, 2 channels per lane.
__global__ __launch_bounds__(256) void k_scatter(const int* __restrict__ ei,
                                                 const float* __restrict__ hs,
                                                 float* __restrict__ acc) {
  int wave = (blockIdx.x * 256 + threadIdx.x) >> 5;
  int lane = threadIdx.x & 31;
  if (wave >= N_EDGES) return;
  int s = ei[wave];             // wave-uniform -> scalarized load
  int d = ei[N_EDGES + wave];
  v2f m = *(const v2f*)(hs + s * CH + lane * 2);
  float* p = acc + d * CH + lane * 2;
  atomicAdd(p,     m.x);
  atomicAdd(p + 1, m.y);
}

// 6) out = relu(dinv[row] * acc + b), float4 vectorized, in-place on d_out.
__global__ __launch_bounds__(256) void k_final(float* __restrict__ out,
                                               const float* __restrict__ dinv,
                                               const float* __restrict__ b) {
  int idx = blockIdx.x * 256 + threadIdx.x;      // N_NODES*16 threads
  if (idx >= N_NODES * 16) return;
  int row = idx >> 4;
  int c0  = (idx & 15) << 2;
  float s = dinv[row];
  float4 v  = *(float4*)(out + row * CH + c0);
  float4 bb = *(const float4*)(b + c0);
  v.x = fmaxf(fmaf(v.x, s, bb.x), 0.0f);
  v.y = fmaxf(fmaf(v.y, s, bb.y), 0.0f);
  v.z = fmaxf(fmaf(v.z, s, bb.z), 0.0f);
  v.w = fmaxf(fmaf(v.w, s, bb.w), 0.0f);
  *(float4*)(out + row * CH + c0) = v;
}

// ---------------------------------------------------------------------------
extern "C" void kernel_launch(void* const* d_in, const int* in_sizes, int n_in,
                              void* d_out, int out_size, void* d_ws, size_t ws_size,
                              hipStream_t stream) {
  const float* x  = (const float*)d_in[0];   // [100000,64] f32
  const int*   ei = (const int*)d_in[1];     // [2,1280000] i32
  const float* W  = (const float*)d_in[2];   // [64,64] f32
  const float* b  = (const float*)d_in[3];   // [64] f32
  float* out = (float*)d_out;                // [100000,64] f32 (also accumulator)

  // Workspace: dinv (N floats) + hs (N*CH floats) ~ 26 MB total.
  float* deg = (float*)d_ws;
  float* hs  = (float*)((char*)d_ws +
                        ((N_NODES * sizeof(float) + 255) & ~(size_t)255));

  k_init_deg <<<(N_NODES + 255) / 256, 256, 0, stream>>>(deg);
  k_count_deg<<<(N_EDGES + 255) / 256, 256, 0, stream>>>(ei, deg);
  k_rsqrt    <<<(N_NODES + 255) / 256, 256, 0, stream>>>(deg);
  // 25000 wave-tiles, 8 waves per 256-thread block -> 3125 blocks (exact)
  k_gemm     <<<(N_NODES / 16) * (CH / 16) / 8, 256, 0, stream>>>(x, W, deg, hs, out);
  // one wave per edge, 8 edges per block -> 160000 blocks (exact)
  k_scatter  <<<N_EDGES / 8, 256, 0, stream>>>(ei, hs, out);
  k_final    <<<(N_NODES * 16 + 255) / 256, 256, 0, stream>>>(out, deg, b);
}